// Grapher_70351564309001
// MI455X (gfx1250) — compile-verified
//
#include <hip/hip_runtime.h>
#include <math.h>

// Problem constants (from the reference)
#define NNODES 16384     // B*H*W = 4*64*64
#define CFEAT  48
#define KNN    9
#define NSLICE 4         // column slices for kNN parallelism
#define COLS_PER_SLICE (NNODES / NSLICE)         // 4096
#define TILES_PER_SLICE (COLS_PER_SLICE / 16)    // 256
#define GLPAD 20         // Gram LDS row stride (floats): 80B -> 16B aligned b128 reads

typedef __attribute__((ext_vector_type(2))) float v2f;
typedef __attribute__((ext_vector_type(8))) float v8f;

__device__ __forceinline__ int batch_of(int n) {
  // floor(linspace(0,4,N)) == floor(4n/16383) computed exactly in ints
  return (n * 4) / 16383;
}

// ---------------------------------------------------------------------------
// 1) Transpose [B,C,H,W] -> xf[N,C]
// ---------------------------------------------------------------------------
__global__ void prep_transpose(const float* __restrict__ x, float* __restrict__ xf) {
  int idx = blockIdx.x * blockDim.x + threadIdx.x;
  if (idx >= NNODES * CFEAT) return;
  int n = idx / CFEAT;
  int c = idx - n * CFEAT;
  int b = n >> 12;          // H*W = 4096
  int hw = n & 4095;
  xf[idx] = x[((b * CFEAT + c) << 12) + hw];
}

// ---------------------------------------------------------------------------
// 2) Squared norms + zero degree counters
// ---------------------------------------------------------------------------
__global__ void prep_sq(const float* __restrict__ xf, float* __restrict__ sq,
                        unsigned* __restrict__ deg) {
  int n = blockIdx.x * blockDim.x + threadIdx.x;
  if (n >= NNODES) return;
  float s = 0.f;
#pragma unroll
  for (int c = 0; c < CFEAT; ++c) {
    float v = xf[n * CFEAT + c];
    s += v * v;
  }
  sq[n] = s;
  deg[n] = 0u;
}

// ---------------------------------------------------------------------------
// Helper: load one 16-col B panel (4x16 f32 WMMA operand per K-step).
// Lane holds N=lane&15, K = 4s + 2*(lane>>4) .. +1  -> one float2 per step.
// ---------------------------------------------------------------------------
__device__ __forceinline__ void load_btile(const float* __restrict__ xf, int colBase,
                                           int lm, int half, v2f (&bm)[12]) {
  const int cn = colBase + lm;
#pragma unroll
  for (int s = 0; s < 12; ++s) {
    int k = 4 * s + 2 * half;
    bm[s] = *(const v2f*)(xf + cn * CFEAT + k);
  }
}

// ---------------------------------------------------------------------------
// 3) kNN: Gram tiles via V_WMMA_F32_16X16X4_F32, per-row top-9 per col-slice.
//    Block = 256 threads = 8 independent waves; each wave owns 32 rows
//    (two 16x16 accumulators). grid = (64, NSLICE).
//    - slice sq[] + batch[] staged in LDS once (single barrier)
//    - B panels double-buffered in registers to hide L2 latency
//    - no per-tile barriers: per-wave LDS Gram staging, DS is in-order per wave
// ---------------------------------------------------------------------------
__launch_bounds__(256)
__global__ void knn_kernel(const float* __restrict__ xf, const float* __restrict__ sq,
                           float* __restrict__ candd, int* __restrict__ candi) {
  __shared__ float GL[8][32][GLPAD];        // per-wave Gram tile (aligned rows)
  __shared__ float sqS[COLS_PER_SLICE];     // slice col squared norms
  __shared__ float batchS[COLS_PER_SLICE];  // slice col batch ids (as float)

  const int lane = threadIdx.x & 31;
  const int wave = threadIdx.x >> 5;
  const int half = lane >> 4;       // which half of the wave (K stripe)
  const int lm   = lane & 15;
  const int rowBase = (blockIdx.x * 8 + wave) * 32;
  const int slice   = blockIdx.y;
  const int sliceBase = slice * COLS_PER_SLICE;

  // Stage slice metadata once; the only block-wide barrier in the kernel.
  for (int i = threadIdx.x; i < COLS_PER_SLICE; i += 256) {
    int col = sliceBase + i;
    sqS[i] = sq[col];
    batchS[i] = (float)batch_of(col);
  }
  __syncthreads();

  // Preload A operands for the two 16-row tiles (rows rowBase .. rowBase+31).
  // A layout (16x4 f32): lane holds M=lane&15, K = 4s + 2*(lane>>4) .. +1
  v2f a0[12], a1[12];
  {
    const int r0 = rowBase + lm;
    const int r1 = rowBase + 16 + lm;
#pragma unroll
    for (int s = 0; s < 12; ++s) {
      int k = 4 * s + 2 * half;
      a0[s] = *(const v2f*)(xf + r0 * CFEAT + k);
      a1[s] = *(const v2f*)(xf + r1 * CFEAT + k);
    }
  }

  const int   myrow = rowBase + lane;   // row this lane scans
  const float sqr   = sq[myrow];
  const float browf = (float)batch_of(myrow);

  float topd[KNN];
  int   topi[KNN];
#pragma unroll
  for (int k = 0; k < KNN; ++k) { topd[k] = INFINITY; topi[k] = k; }

  // Software-pipelined B panel (double buffer in registers).
  v2f bcur[12], bnext[12];
  load_btile(xf, sliceBase, lm, half, bcur);

  for (int t = 0; t < TILES_PER_SLICE; ++t) {
    // Issue next panel's loads before this tile's WMMA chain.
    if (t + 1 < TILES_PER_SLICE)
      load_btile(xf, sliceBase + (t + 1) * 16, lm, half, bnext);

    v8f acc0 = {0.f, 0.f, 0.f, 0.f, 0.f, 0.f, 0.f, 0.f};
    v8f acc1 = {0.f, 0.f, 0.f, 0.f, 0.f, 0.f, 0.f, 0.f};
#pragma unroll
    for (int s = 0; s < 12; ++s) {
      acc0 = __builtin_amdgcn_wmma_f32_16x16x4_f32(false, a0[s], false, bcur[s],
                                                   (short)0, acc0, false, false);
      acc1 = __builtin_amdgcn_wmma_f32_16x16x4_f32(false, a1[s], false, bcur[s],
                                                   (short)0, acc1, false, false);
    }

    // Spill Gram tile to per-wave LDS: C/D layout M = v + 8*(lane>=16), N = lane&15.
    // DS ops are in-order within a wave -> no barrier needed.
#pragma unroll
    for (int v = 0; v < 8; ++v) {
      GL[wave][v + 8 * half][lm]      = acc0[v];
      GL[wave][16 + v + 8 * half][lm] = acc1[v];
    }

    // Scan: lane r owns local row r; 16 candidates from this tile.
    const int lbase = t * 16;
#pragma unroll
    for (int j = 0; j < 16; ++j) {
      float d = sqr + sqS[lbase + j] - 2.0f * GL[wave][lane][j];
      if (batchS[lbase + j] == browf && d < topd[KNN - 1]) {
        topd[KNN - 1] = d;
        topi[KNN - 1] = sliceBase + lbase + j;
#pragma unroll
        for (int k = KNN - 1; k > 0; --k) {
          if (topd[k] < topd[k - 1]) {
            float td = topd[k]; topd[k] = topd[k - 1]; topd[k - 1] = td;
            int   ti = topi[k]; topi[k] = topi[k - 1]; topi[k - 1] = ti;
          }
        }
      }
    }

    // Rotate double buffer.
#pragma unroll
    for (int s = 0; s < 12; ++s) bcur[s] = bnext[s];
  }

  // Emit per-slice candidates (sorted ascending by distance).
  const size_t base = ((size_t)slice * NNODES + myrow) * KNN;
#pragma unroll
  for (int k = 0; k < KNN; ++k) {
    candd[base + k] = topd[k];
    candi[base + k] = topi[k];
  }
}

// ---------------------------------------------------------------------------
// 4) Merge NSLICE partial top-9 lists -> nbr[N,9]; accumulate degrees.
// ---------------------------------------------------------------------------
__global__ void merge_kernel(const float* __restrict__ candd, const int* __restrict__ candi,
                             int* __restrict__ nbr, unsigned* __restrict__ deg) {
  int n = blockIdx.x * blockDim.x + threadIdx.x;
  if (n >= NNODES) return;
  float topd[KNN];
  int   topi[KNN];
#pragma unroll
  for (int k = 0; k < KNN; ++k) { topd[k] = INFINITY; topi[k] = k; }
  for (int s = 0; s < NSLICE; ++s) {
    const size_t base = ((size_t)s * NNODES + n) * KNN;
#pragma unroll
    for (int c = 0; c < KNN; ++c) {
      float d = candd[base + c];
      int  ci = candi[base + c];
      if (d < topd[KNN - 1]) {
        topd[KNN - 1] = d;
        topi[KNN - 1] = ci;
#pragma unroll
        for (int k = KNN - 1; k > 0; --k) {
          if (topd[k] < topd[k - 1]) {
            float td = topd[k]; topd[k] = topd[k - 1]; topd[k - 1] = td;
            int   ti = topi[k]; topi[k] = topi[k - 1]; topi[k - 1] = ti;
          }
        }
      }
    }
  }
#pragma unroll
  for (int k = 0; k < KNN; ++k) {
    nbr[n * KNN + k] = topi[k];
    atomicAdd(&deg[topi[k]], 1u);
  }
}

// ---------------------------------------------------------------------------
// 5a) dinv = rsqrt(max(deg,1)) (deg>0 guard as in reference)
// ---------------------------------------------------------------------------
__global__ void dinv_kernel(const unsigned* __restrict__ deg, float* __restrict__ dinv) {
  int n = blockIdx.x * blockDim.x + threadIdx.x;
  if (n >= NNODES) return;
  float d = (float)deg[n];
  dinv[n] = (d > 0.f) ? rsqrtf(fmaxf(d, 1.f)) : 0.f;
}

// ---------------------------------------------------------------------------
// 5b) tx1[i,c] = -dinv[i] * sum_k dinv[nbr[i,k]] * xf[nbr[i,k], c]
// ---------------------------------------------------------------------------
__global__ void tx1_kernel(const float* __restrict__ xf, const int* __restrict__ nbr,
                           const float* __restrict__ dinv, float* __restrict__ tx1) {
  int idx = blockIdx.x * blockDim.x + threadIdx.x;
  if (idx >= NNODES * CFEAT) return;
  int i = idx / CFEAT;
  int c = idx - i * CFEAT;
  float s = 0.f;
#pragma unroll
  for (int k = 0; k < KNN; ++k) {
    int j = nbr[i * KNN + k];
    s += dinv[j] * xf[j * CFEAT + c];
  }
  tx1[idx] = -dinv[i] * s;
}

// ---------------------------------------------------------------------------
// 6) out = relu(xf@W0 + tx1@W1 + b) via f32 WMMA.
//    Block = 256 threads = 8 waves; each wave does 16 rows x all 48 out cols.
// ---------------------------------------------------------------------------
__launch_bounds__(256)
__global__ void out_kernel(const float* __restrict__ xf, const float* __restrict__ tx1,
                           const float* __restrict__ W0, const float* __restrict__ W1,
                           const float* __restrict__ bias, float* __restrict__ out) {
  const int lane = threadIdx.x & 31;
  const int wave = threadIdx.x >> 5;
  const int half = lane >> 4;
  const int lm   = lane & 15;
  const int rowBase = (blockIdx.x * 8 + wave) * 16;
  const int r = rowBase + lm;

  v2f ax[12], at[12];
#pragma unroll
  for (int s = 0; s < 12; ++s) {
    int k = 4 * s + 2 * half;
    ax[s] = *(const v2f*)(xf + r * CFEAT + k);
    at[s] = *(const v2f*)(tx1 + r * CFEAT + k);
  }

  for (int t = 0; t < 3; ++t) {            // 3 x 16 output columns
    const int colBase = t * 16;
    v8f acc = {0.f, 0.f, 0.f, 0.f, 0.f, 0.f, 0.f, 0.f};
#pragma unroll
    for (int s = 0; s < 12; ++s) {
      int k = 4 * s + 2 * half;
      v2f b0;
      b0[0] = W0[k * CFEAT + colBase + lm];
      b0[1] = W0[(k + 1) * CFEAT + colBase + lm];
      acc = __builtin_amdgcn_wmma_f32_16x16x4_f32(false, ax[s], false, b0,
                                                  (short)0, acc, false, false);
    }
#pragma unroll
    for (int s = 0; s < 12; ++s) {
      int k = 4 * s + 2 * half;
      v2f b1;
      b1[0] = W1[k * CFEAT + colBase + lm];
      b1[1] = W1[(k + 1) * CFEAT + colBase + lm];
      acc = __builtin_amdgcn_wmma_f32_16x16x4_f32(false, at[s], false, b1,
                                                  (short)0, acc, false, false);
    }
    float bb = bias[colBase + lm];
#pragma unroll
    for (int v = 0; v < 8; ++v) {
      int row = rowBase + v + 8 * half;
      out[row * CFEAT + colBase + lm] = fmaxf(acc[v] + bb, 0.f);
    }
  }
}

// ---------------------------------------------------------------------------
// Launch
// ---------------------------------------------------------------------------
extern "C" void kernel_launch(void* const* d_in, const int* in_sizes, int n_in,
                              void* d_out, int out_size, void* d_ws, size_t ws_size,
                              hipStream_t stream) {
  const float* x  = (const float*)d_in[0];
  const float* W0 = (const float*)d_in[1];
  const float* W1 = (const float*)d_in[2];
  const float* b  = (const float*)d_in[3];
  float* out = (float*)d_out;

  // Carve workspace (all sizes in bytes). Total ~11.8 MB.
  char* ws = (char*)d_ws;
  float*    xf    = (float*)ws;    ws += (size_t)NNODES * CFEAT * 4;
  float*    sq    = (float*)ws;    ws += (size_t)NNODES * 4;
  unsigned* deg   = (unsigned*)ws; ws += (size_t)NNODES * 4;
  float*    dinv  = (float*)ws;    ws += (size_t)NNODES * 4;
  float*    tx1   = (float*)ws;    ws += (size_t)NNODES * CFEAT * 4;
  int*      nbr   = (int*)ws;      ws += (size_t)NNODES * KNN * 4;
  float*    candd = (float*)ws;    ws += (size_t)NSLICE * NNODES * KNN * 4;
  int*      candi = (int*)ws;      ws += (size_t)NSLICE * NNODES * KNN * 4;

  prep_transpose<<<(NNODES * CFEAT + 255) / 256, 256, 0, stream>>>(x, xf);
  prep_sq<<<(NNODES + 255) / 256, 256, 0, stream>>>(xf, sq, deg);

  dim3 kg(NNODES / (8 * 32), NSLICE);   // 64 x 4 blocks
  knn_kernel<<<kg, 256, 0, stream>>>(xf, sq, candd, candi);

  merge_kernel<<<(NNODES + 255) / 256, 256, 0, stream>>>(candd, candi, nbr, deg);
  dinv_kernel<<<(NNODES + 255) / 256, 256, 0, stream>>>(deg, dinv);
  tx1_kernel<<<(NNODES * CFEAT + 255) / 256, 256, 0, stream>>>(xf, nbr, dinv, tx1);
  out_kernel<<<NNODES / (8 * 16), 256, 0, stream>>>(xf, tx1, W0, W1, b, out);
}